// TTLinear_10531259810490
// MI455X (gfx1250) — compile-verified
//
#include <hip/hip_runtime.h>

// TT-linear, rank-8 factored:  out = (x @ core1^T) @ core0^T + bias
// Fused single kernel, minimal HBM traffic (read x once NT, write out once NT).
// Matrix ops on the CDNA5 matrix pipe via V_WMMA_F32_16X16X4_F32.

typedef __attribute__((ext_vector_type(2))) float v2f;
typedef __attribute__((ext_vector_type(8))) float v8f;

#define IN_F  4096
#define OUT_F 4096
#define RANK  8
#define TM    16   // tokens per workgroup

__global__ __launch_bounds__(256) void tt_linear_fused(
    const float* __restrict__ x,      // [TOKENS, IN_F]
    const float* __restrict__ core0,  // [OUT_F, RANK]   (core0[0,o,r] -> o*8+r)
    const float* __restrict__ core1,  // [RANK, IN_F]    (core1[r,i,0] -> r*4096+i)
    const float* __restrict__ bias,   // [OUT_F]
    float* __restrict__ out)          // [TOKENS, OUT_F]
{
    __shared__ float lds_t[TM * 16];  // t tile, N padded 8 -> 16

    const int tid   = threadIdx.x;
    const int lane  = tid & 31;
    const int wave  = tid >> 5;       // 0..7
    const int lidx  = lane & 15;      // A-frag M index == B-frag N index == C N index
    const int khalf = lane >> 4;      // 0 or 1 (K split across lane halves)
    const int kb    = khalf * 2;      // K sub-offset within a K=4 chunk
    const int tok0  = blockIdx.x * TM;

    // zero-init the t reduction tile (256 threads cover 256 floats)
    lds_t[tid] = 0.0f;
    __syncthreads();

    // ---------------- Stage A: t[16 x 8] = x_tile[16 x 4096] @ core1^T ---------
    // A fragment (16x4 f32): lane = token row, {VGPR0,VGPR1} = K pair, halves = K+2
    // B fragment (4x16 f32): lane = rank column (only 0..7 valid -> zero mask)
    const float nmask = (lidx < RANK) ? 1.0f : 0.0f;
    const float* xrow  = x + (size_t)(tok0 + lidx) * IN_F;
    const float* c1row = core1 + (size_t)(lidx & (RANK - 1)) * IN_F;

    v8f acc = {};
    const int kbeg = wave * (IN_F / 8);   // each wave: contiguous K range of 512
    const int kend = kbeg + (IN_F / 8);
    for (int k0 = kbeg; k0 < kend; k0 += 32) {
        // one GL2 streaming prefetch per 128B-per-lane of x (speculative; OOB
        // translation failures at the tail are silently dropped per ISA)
        __builtin_prefetch(xrow + k0 + kb + 1024, 0, 0);
#pragma unroll
        for (int kk = 0; kk < 32; kk += 4) {
            const int k = k0 + kk;
            // x is streamed exactly once -> non-temporal load (TH=NT)
            v2f a = __builtin_nontemporal_load((const v2f*)(xrow + k + kb));
            v2f b = *(const v2f*)(c1row + k + kb);    // hot in L1/L2 (128 KB)
            b.x *= nmask;
            b.y *= nmask;
            acc = __builtin_amdgcn_wmma_f32_16x16x4_f32(
                /*neg_a=*/false, a, /*neg_b=*/false, b,
                /*c_mod=*/(short)0, acc, /*reuse_a=*/false, /*reuse_b=*/false);
        }
    }

    // cross-wave reduction of the 16x16 partial tiles (ds_add_f32)
#pragma unroll
    for (int v = 0; v < 8; ++v) {
        atomicAdd(&lds_t[(v + 8 * khalf) * 16 + lidx], acc[v]);
    }
    __syncthreads();

    // ---------------- Stage B: out_tile = t[16 x 8] @ core0^T + bias -----------
    // t A-fragments (same for all output tiles): two K=4 chunks
    v2f ta0, ta1;
    ta0.x = lds_t[lidx * 16 + kb + 0];
    ta0.y = lds_t[lidx * 16 + kb + 1];
    ta1.x = lds_t[lidx * 16 + 4 + kb + 0];
    ta1.y = lds_t[lidx * 16 + 4 + kb + 1];

    float* const orow0 = out + (size_t)(tok0 + 8 * khalf) * OUT_F;

    for (int ot = wave; ot < OUT_F / 16; ot += 8) {
        const int o = ot * 16 + lidx;            // this lane's output column
        const float bval = bias[o];
        v2f b0 = *(const v2f*)(core0 + o * RANK + kb);       // K = 0..3
        v2f b1 = *(const v2f*)(core0 + o * RANK + 4 + kb);   // K = 4..7

        v8f c;
#pragma unroll
        for (int v = 0; v < 8; ++v) c[v] = bval;  // bias pre-loaded into C

        c = __builtin_amdgcn_wmma_f32_16x16x4_f32(false, ta0, false, b0,
                                                  (short)0, c, false, false);
        c = __builtin_amdgcn_wmma_f32_16x16x4_f32(false, ta1, false, b1,
                                                  (short)0, c, false, false);

        // C layout: lane holds column o for rows (v + 8*khalf).
        // out is written exactly once -> non-temporal store (TH=NT).
#pragma unroll
        for (int v = 0; v < 8; ++v) {
            __builtin_nontemporal_store(c[v], &orow0[(size_t)v * OUT_F + o]);
        }
    }
}

extern "C" void kernel_launch(void* const* d_in, const int* in_sizes, int n_in,
                              void* d_out, int out_size, void* d_ws, size_t ws_size,
                              hipStream_t stream) {
    (void)n_in; (void)out_size; (void)d_ws; (void)ws_size;
    const float* x     = (const float*)d_in[0];
    const float* core0 = (const float*)d_in[1];
    const float* core1 = (const float*)d_in[2];
    const float* bias  = (const float*)d_in[3];
    float* out = (float*)d_out;

    const int tokens = in_sizes[0] / IN_F;   // 8192
    const int grid   = tokens / TM;          // 512 workgroups
    tt_linear_fused<<<grid, 256, 0, stream>>>(x, core0, core1, bias, out);
}